// PSRoIMaskPool_76871324664412
// MI455X (gfx1250) — compile-verified
//
#include <hip/hip_runtime.h>

typedef float v2f __attribute__((ext_vector_type(2)));
typedef float v8f __attribute__((ext_vector_type(8)));
typedef int   v4i __attribute__((vector_size(16)));  // matches builtin param type

#define Hdim 128
#define Wdim 128
#define Cdim 490
#define Ddim 10
#define LDSS 132 // padded LDS row stride in floats (528B, 16B aligned, conflict-avoiding)

#if defined(__AMDGCN__) && __has_builtin(__builtin_amdgcn_global_load_async_to_lds_b128) && \
    __has_builtin(__builtin_amdgcn_s_wait_asynccnt)
#define USE_ASYNC_LDS 1
typedef __attribute__((address_space(1))) v4i* gptr4;  // global (AS1) int4*
typedef __attribute__((address_space(3))) v4i* lptr4;  // LDS (AS3) int4*
#else
#define USE_ASYNC_LDS 0
#endif

// ---------------- Phase 1: per-plane 2D inclusive prefix sum (SAT) via f32 WMMA ----------------
// cumsum along W:  out = X @ U16 per 16-col tile + running row-sum carry (U upper-tri incl.)
// cumsum along H:  out = L16 @ Y per 16-row tile + running col-sum carry (L lower-tri incl.)
__global__ __launch_bounds__(256) void sat_kernel(const float* __restrict__ feat,
                                                  float* __restrict__ cs) {
  __shared__ __align__(16) float tile[Hdim * LDSS];
  const int tid  = threadIdx.x;
  const int lane = tid & 31;
  const int wave = tid >> 5;
  const size_t planeBase = (size_t)blockIdx.x * (size_t)(Hdim * Wdim);
  const float* src = feat + planeBase;

#if USE_ASYNC_LDS
  // CDNA5 async DMA: global -> LDS without VGPR staging (tracked by ASYNCcnt).
  for (int v = tid; v < (Hdim * Wdim / 4); v += 256) {
    int f = v << 2;
    int row = f >> 7;          // /128
    int col = f & (Wdim - 1);  // %128
    __builtin_amdgcn_global_load_async_to_lds_b128(
        (gptr4)(float*)(src + f),
        (lptr4)&tile[row * LDSS + col],
        0, 0);
  }
  __builtin_amdgcn_s_wait_asynccnt(0);
#else
  for (int v = tid; v < (Hdim * Wdim / 4); v += 256) {
    float4 d = ((const float4*)src)[v];
    int f = v << 2;
    int row = f >> 7;
    int col = f & (Wdim - 1);
    *(float4*)(&tile[row * LDSS + col]) = d;
  }
#endif
  __syncthreads();

  const int half = lane >> 4;  // 0: lanes 0-15, 1: lanes 16-31
  const int m    = lane & 15;  // M (or N) index within tile for this lane

  // ---------- Row-direction cumsum: wave owns rows [rb, rb+16) ----------
  {
    const int rb = wave * 16;
    v8f carry;
    for (int r = 0; r < 8; ++r) carry[r] = 0.0f;
    for (int jb = 0; jb < 8; ++jb) {
      v8f acc = carry;  // C[m][n] = rowcarry[m] broadcast across N
      for (int kc = 0; kc < 4; ++kc) {
        const int subk = half << 1;        // K sub-index 0/2 per lane half
        const int k0   = kc * 4 + subk;    // K within 16-wide tile
        // A = X tile slice 16x4 (fp32): lane holds X[rb+m][koff], X[rb+m][koff+1]
        v2f a = *(const v2f*)(&tile[(rb + m) * LDSS + jb * 16 + k0]);
        // B = U16 chunk 4x16: U[k][n] = (k <= n)
        v2f b;
        b.x = (k0     <= m) ? 1.0f : 0.0f;
        b.y = (k0 + 1 <= m) ? 1.0f : 0.0f;
        acc = __builtin_amdgcn_wmma_f32_16x16x4_f32(false, a, false, b,
                                                    (short)0, acc, false, false);
      }
      // New carry = column N=15 of D (lane 15 / lane 31 per half).
      const int srcl = (lane & 16) | 15;
      v8f nc;
      for (int r = 0; r < 8; ++r) nc[r] = __shfl(acc[r], srcl, 32);
      carry = nc;
      // Store D tile back over X tile (all reads of this tile already done).
      const int mb = rb + (half << 3);
      const int nn = jb * 16 + m;
      for (int r = 0; r < 8; ++r) tile[(mb + r) * LDSS + nn] = acc[r];
    }
  }
  __syncthreads();

  // ---------- Column-direction cumsum: wave owns cols [cb, cb+16); write SAT to global ----------
  {
    const int cb = wave * 16;
    float ccar = 0.0f;  // running column sum for column cb+m (per lane)
    float* dst = cs + planeBase;
    for (int ib = 0; ib < 8; ++ib) {
      v8f acc;
      for (int r = 0; r < 8; ++r) acc[r] = ccar;  // C[m][n] = colcarry[n]
      for (int kc = 0; kc < 4; ++kc) {
        const int subk = half << 1;
        const int k0   = kc * 4 + subk;
        // A = L16 chunk 16x4: L[m][k] = (k <= m)
        v2f a;
        a.x = (k0     <= m) ? 1.0f : 0.0f;
        a.y = (k0 + 1 <= m) ? 1.0f : 0.0f;
        // B = Y tile rows k0,k0+1 (cols cb..cb+15)
        const int krow = ib * 16 + k0;
        v2f b;
        b.x = tile[krow * LDSS + cb + m];
        b.y = tile[(krow + 1) * LDSS + cb + m];
        acc = __builtin_amdgcn_wmma_f32_16x16x4_f32(false, a, false, b,
                                                    (short)0, acc, false, false);
      }
      // New column carry = row M=15 of D: lives at lane 16+n, VGPR 7.
      ccar = __shfl(acc[7], 16 + m, 32);
      const int rowb = ib * 16 + (half << 3);
      for (int r = 0; r < 8; ++r)
        dst[(size_t)(rowb + r) * Wdim + cb + m] = acc[r];
    }
  }
}

// ---------------- Phase 2: PS-RoI pooled gather from SAT (served from L2) ----------------
__global__ __launch_bounds__(256) void pool_kernel(const float* __restrict__ rois,
                                                   const float* __restrict__ cs,
                                                   const int* __restrict__ stridep,
                                                   float* __restrict__ out, int total) {
  int idx = blockIdx.x * blockDim.x + threadIdx.x;
  if (idx >= total) return;
  int pj = idx % 7;
  int t  = idx / 7;
  int pi = t % 7;
  t /= 7;
  int d = t % Ddim;
  int n = t / Ddim;

  const float* roi = rois + n * 5;
  const float s = 1.0f / (float)stridep[0];
  const int bi = (int)roi[0];
  float rsw = rintf(roi[1]) * s;           // jnp.round == RNE == rintf
  float rsh = rintf(roi[2]) * s;
  float rew = (rintf(roi[3]) + 1.0f) * s;
  float reh = (rintf(roi[4]) + 1.0f) * s;
  float rwm = (rew - rsw) * 1.3f;
  float rhm = (reh - rsh) * 1.3f;
  float swm = (rsw + rew) * 0.5f - rwm * 0.5f;
  float shm = (rsh + reh) * 0.5f - rhm * 0.5f;
  rwm = fmaxf(rwm, 0.1f);
  rhm = fmaxf(rhm, 0.1f);
  float bin_h = rhm / 7.0f;
  float bin_w = rwm / 7.0f;
  float dh = bin_h * 0.25f;  // (BIN_SCALE-1)*0.5
  float dw = bin_w * 0.25f;

  int hs = (int)fminf(fmaxf(floorf(shm + (float)pi * bin_h - dh), 0.0f), 128.0f);
  int he = (int)fminf(fmaxf(ceilf (shm + (float)(pi + 1) * bin_h + dh), 0.0f), 128.0f);
  int ws = (int)fminf(fmaxf(floorf(swm + (float)pj * bin_w - dw), 0.0f), 128.0f);
  int we = (int)fminf(fmaxf(ceilf (swm + (float)(pj + 1) * bin_w + dw), 0.0f), 128.0f);

  const int c = d * 49 + pi * 7 + pj;  // Sr = S.reshape(B, D, G, G, ...)
  const float* base = cs + (size_t)(bi * Cdim + c) * (size_t)(Hdim * Wdim);
  auto S = [&](int i, int j) -> float {
    return (i > 0 && j > 0) ? base[(size_t)(i - 1) * Wdim + (j - 1)] : 0.0f;
  };
  float tv = S(he, we) - S(hs, we) - S(he, ws) + S(hs, ws);
  int area = (he - hs) * (we - ws);
  out[idx] = (area > 0) ? tv / (float)area : 0.0f;
}

extern "C" void kernel_launch(void* const* d_in, const int* in_sizes, int n_in,
                              void* d_out, int out_size, void* d_ws, size_t ws_size,
                              hipStream_t stream) {
  const float* rois    = (const float*)d_in[0];
  const float* feat    = (const float*)d_in[1];
  const int*   stridep = (const int*)d_in[2];
  float* cs  = (float*)d_ws;   // SAT scratch: 980 * 128 * 128 * 4B = 64.2 MB
  float* out = (float*)d_out;

  const int planes = in_sizes[1] / (Hdim * Wdim);  // B*C = 980
  sat_kernel<<<planes, 256, 0, stream>>>(feat, cs);
  pool_kernel<<<(out_size + 255) / 256, 256, 0, stream>>>(rois, cs, stridep, out, out_size);
}